// PeriodicDistance_59064390254854
// MI455X (gfx1250) — compile-verified
//
#include <hip/hip_runtime.h>
#include <hip/hip_bf16.h>

// Periodic distance gather (memory-bound). Output layout in d_out (fp32):
//   [0      , 2E)  edge_index (as float, row0 then row1)
//   [2E     , 3E)  edge_weight
//   [3E     , 6E)  edge_vec   (flattened [E,3])
//   [6E     , 9E)  shifts_idx (as float, flattened [E,3])

#define PF_EDGES 16384  // prefetch distance (edges ahead) for streaming arrays

__global__ __launch_bounds__(256) void PeriodicDistance_kernel(
    const float* __restrict__ pos,     // [N,3]
    const float* __restrict__ box,     // [B,3,3]
    const int*   __restrict__ ei,      // [2,E]
    const int*   __restrict__ shifts,  // [E,3]
    const int*   __restrict__ batch,   // [E]
    float* __restrict__ out,
    int E)
{
    const int q  = blockIdx.x * blockDim.x + threadIdx.x;  // quad index
    const int e0 = q * 4;
    if (e0 >= E) return;

    const long long Ell = (long long)E;
    float* out_ei0 = out;                 // [E]
    float* out_ei1 = out + Ell;           // [E]
    float* out_w   = out + 2 * Ell;       // [E]
    float* out_vec = out + 3 * Ell;       // [E,3]
    float* out_sh  = out + 6 * Ell;       // [E,3]

    // gfx1250 WGP-scope prefetch (locality=3 -> fill all cache levels on miss)
    // of the streaming arrays well ahead of this wave's work; hides latency
    // behind the random pos gathers. No dependency counters are consumed.
    if (e0 + PF_EDGES < E) {
        __builtin_prefetch(ei + e0 + PF_EDGES, 0, 3);
        __builtin_prefetch(ei + Ell + e0 + PF_EDGES, 0, 3);
        __builtin_prefetch(shifts + (long long)(e0 + PF_EDGES) * 3, 0, 3);
        __builtin_prefetch(batch + e0 + PF_EDGES, 0, 3);
    }

    if (e0 + 3 < E) {
        // ---------------- vector (quad) path: all streaming traffic is b128 ----
        const int4 src = *reinterpret_cast<const int4*>(ei + e0);        // row 0
        const int4 dst = *reinterpret_cast<const int4*>(ei + Ell + e0);  // row 1
        const int4 sA  = *reinterpret_cast<const int4*>(shifts + (long long)e0 * 3 + 0);
        const int4 sB  = *reinterpret_cast<const int4*>(shifts + (long long)e0 * 3 + 4);
        const int4 sC  = *reinterpret_cast<const int4*>(shifts + (long long)e0 * 3 + 8);
        const int4 bm  = *reinterpret_cast<const int4*>(batch + e0);

        int sh[12];
        sh[0] = sA.x; sh[1]  = sA.y; sh[2]  = sA.z; sh[3]  = sA.w;
        sh[4] = sB.x; sh[5]  = sB.y; sh[6]  = sB.z; sh[7]  = sB.w;
        sh[8] = sC.x; sh[9]  = sC.y; sh[10] = sC.z; sh[11] = sC.w;
        int i0v[4] = {src.x, src.y, src.z, src.w};
        int i1v[4] = {dst.x, dst.y, dst.z, dst.w};
        int bv[4]  = {bm.x, bm.y, bm.z, bm.w};

        float wq[4], vecq[12], shf[12], ef0[4], ef1[4];

#pragma unroll
        for (int k = 0; k < 4; ++k) {
            const int   i0 = i0v[k];
            const int   i1 = i1v[k];
            const float sx = (float)sh[3 * k + 0];
            const float sy = (float)sh[3 * k + 1];
            const float sz = (float)sh[3 * k + 2];

            const float* B = box + (long long)bv[k] * 9;  // tiny, cache resident
            // cell_shift_m = s0*B[0,m] + s1*B[1,m] + s2*B[2,m]
            const float csx = fmaf(sx, B[0], fmaf(sy, B[3], sz * B[6]));
            const float csy = fmaf(sx, B[1], fmaf(sy, B[4], sz * B[7]));
            const float csz = fmaf(sx, B[2], fmaf(sy, B[5], sz * B[8]));

            const float* p0 = pos + (long long)i0 * 3;  // random gather (L2-resident)
            const float* p1 = pos + (long long)i1 * 3;
            const float drx = p1[0] - p0[0] + csx;
            const float dry = p1[1] - p0[1] + csy;
            const float drz = p1[2] - p0[2] + csz;

            wq[k] = sqrtf(fmaf(drx, drx, fmaf(dry, dry, drz * drz)));
            vecq[3 * k + 0] = -drx;
            vecq[3 * k + 1] = -dry;
            vecq[3 * k + 2] = -drz;
            shf[3 * k + 0] = sx;
            shf[3 * k + 1] = sy;
            shf[3 * k + 2] = sz;
            ef0[k] = (float)i0;   // exact: indices < 2^24
            ef1[k] = (float)i1;
        }

        // ---------------- b128 stores ----------------
        *reinterpret_cast<float4*>(out_ei0 + e0) = make_float4(ef0[0], ef0[1], ef0[2], ef0[3]);
        *reinterpret_cast<float4*>(out_ei1 + e0) = make_float4(ef1[0], ef1[1], ef1[2], ef1[3]);
        *reinterpret_cast<float4*>(out_w + e0)   = make_float4(wq[0], wq[1], wq[2], wq[3]);

        float4* vdst = reinterpret_cast<float4*>(out_vec + (long long)e0 * 3);
        vdst[0] = make_float4(vecq[0], vecq[1], vecq[2],  vecq[3]);
        vdst[1] = make_float4(vecq[4], vecq[5], vecq[6],  vecq[7]);
        vdst[2] = make_float4(vecq[8], vecq[9], vecq[10], vecq[11]);

        float4* sdst = reinterpret_cast<float4*>(out_sh + (long long)e0 * 3);
        sdst[0] = make_float4(shf[0], shf[1], shf[2],  shf[3]);
        sdst[1] = make_float4(shf[4], shf[5], shf[6],  shf[7]);
        sdst[2] = make_float4(shf[8], shf[9], shf[10], shf[11]);
    } else {
        // ---------------- scalar tail (E not multiple of 4) ----------------
        for (int e = e0; e < E; ++e) {
            const int i0 = ei[e];
            const int i1 = ei[Ell + e];
            const float sx = (float)shifts[(long long)e * 3 + 0];
            const float sy = (float)shifts[(long long)e * 3 + 1];
            const float sz = (float)shifts[(long long)e * 3 + 2];
            const float* B = box + (long long)batch[e] * 9;

            const float csx = fmaf(sx, B[0], fmaf(sy, B[3], sz * B[6]));
            const float csy = fmaf(sx, B[1], fmaf(sy, B[4], sz * B[7]));
            const float csz = fmaf(sx, B[2], fmaf(sy, B[5], sz * B[8]));

            const float* p0 = pos + (long long)i0 * 3;
            const float* p1 = pos + (long long)i1 * 3;
            const float drx = p1[0] - p0[0] + csx;
            const float dry = p1[1] - p0[1] + csy;
            const float drz = p1[2] - p0[2] + csz;

            out_ei0[e] = (float)i0;
            out_ei1[e] = (float)i1;
            out_w[e]   = sqrtf(fmaf(drx, drx, fmaf(dry, dry, drz * drz)));
            out_vec[(long long)e * 3 + 0] = -drx;
            out_vec[(long long)e * 3 + 1] = -dry;
            out_vec[(long long)e * 3 + 2] = -drz;
            out_sh[(long long)e * 3 + 0] = sx;
            out_sh[(long long)e * 3 + 1] = sy;
            out_sh[(long long)e * 3 + 2] = sz;
        }
    }
}

extern "C" void kernel_launch(void* const* d_in, const int* in_sizes, int n_in,
                              void* d_out, int out_size, void* d_ws, size_t ws_size,
                              hipStream_t stream) {
    const float* pos    = (const float*)d_in[0];  // [N,3] fp32
    const float* box    = (const float*)d_in[1];  // [B,3,3] fp32
    const int*   ei     = (const int*)d_in[2];    // [2,E] int32
    const int*   shifts = (const int*)d_in[3];    // [E,3] int32
    const int*   batch  = (const int*)d_in[4];    // [E] int32
    float*       out    = (float*)d_out;

    const int E = in_sizes[2] / 2;
    const int quads = (E + 3) / 4;
    const int block = 256;                 // 8 wave32 waves per block
    const int grid  = (quads + block - 1) / block;

    PeriodicDistance_kernel<<<grid, block, 0, stream>>>(pos, box, ei, shifts,
                                                        batch, out, E);
}